// LightGCN_85478439125818
// MI455X (gfx1250) — compile-verified
//
#include <hip/hip_runtime.h>

// ---------------------------------------------------------------------------
// LightGCN propagation on MI455X (gfx1250, wave32).
// mean over {ego, A·ego, A²·ego, A³·ego}; A given as COO (rows, cols, vals).
//
// Memory-bound (~3.5 GB traffic, ~150us HBM floor); the per-edge scaling
// D = diag(val) * X_gathered runs on the matrix pipe via chained
// V_WMMA_F32_16X16X4_F32, whose C/D layout feeds the fp32 atomic scatter
// directly. Gather/scatter base pointers and the diag-A slices are hoisted
// out of the dim-chunk loop so per-chunk work is loads + WMMA + atomics.
// ---------------------------------------------------------------------------

#define NUM_USER 262144
#define NUM_ITEM 262144
#define NTOT     (NUM_USER + NUM_ITEM)
#define DIM      64
#define NNZ_     1250000
#define NLAYERS  3

typedef __attribute__((ext_vector_type(2))) float v2f;
typedef __attribute__((ext_vector_type(4))) float v4f;   // ext-vector: valid for NT builtins
typedef __attribute__((ext_vector_type(8))) float v8f;

// cur = [user;item]; next = 0; out = 0.25 * ego   (b128 path)
__global__ void __launch_bounds__(256) lgcn_init_kernel(
    const float* __restrict__ user, const float* __restrict__ item,
    float* __restrict__ cur, float* __restrict__ next, float* __restrict__ out)
{
    size_t i = (size_t)blockIdx.x * blockDim.x + threadIdx.x;   // float4 units
    const size_t totalv4 = (size_t)NTOT * DIM / 4;
    if (i >= totalv4) return;
    const size_t userv4 = (size_t)NUM_USER * DIM / 4;
    v4f v = (i < userv4) ? ((const v4f*)user)[i]
                         : ((const v4f*)item)[i - userv4];
    ((v4f*)cur)[i] = v;
    v4f z = {0.f, 0.f, 0.f, 0.f};
    __builtin_nontemporal_store(z, (v4f*)next + i);
    ((v4f*)out)[i] = 0.25f * v;
}

// out += 0.25 * next;  zbuf = 0  (zbuf becomes the next scatter target)
__global__ void __launch_bounds__(256) lgcn_addz_kernel(
    float* __restrict__ out, const float* __restrict__ next, float* __restrict__ zbuf)
{
    size_t i = (size_t)blockIdx.x * blockDim.x + threadIdx.x;   // float4 units
    const size_t totalv4 = (size_t)NTOT * DIM / 4;
    if (i >= totalv4) return;
    v4f o = ((v4f*)out)[i];
    v4f n = ((const v4f*)next)[i];
    ((v4f*)out)[i] = o + 0.25f * n;
    v4f z = {0.f, 0.f, 0.f, 0.f};
    __builtin_nontemporal_store(z, (v4f*)zbuf + i);
}

// One wave = 16 edges. For each 16-dim chunk dc:
//   D(16edges x 16dims) = sum_blk A_blk(16x4 slice of diag(val)) x B_blk(4x16 of gathered rows)
// then scatter D with global_atomic_add_f32 (D layout: VGPR r <-> edge r(+8), lane <-> dim).
__global__ void __launch_bounds__(256) lgcn_spmm_wmma_kernel(
    const int*   __restrict__ rows, const int* __restrict__ cols,
    const float* __restrict__ vals,
    const float* __restrict__ x, float* __restrict__ y, int nnz)
{
    const int lane = threadIdx.x & 31;
    const int wave = blockIdx.x * (blockDim.x >> 5) + (threadIdx.x >> 5);
    const int ngroups = (nnz + 15) >> 4;
    if (wave >= ngroups) return;           // wave-uniform: EXEC stays all-ones

    const int  base = wave << 4;
    const int  m    = lane & 15;           // edge slot (lo half) / tile column
    const bool hi   = lane >= 16;

    // per-lane edge data (lanes 16..31 mirror 0..15); pad edges get val = 0
    const int  eraw = base + m;
    const bool live = eraw < nnz;
    const int  e    = live ? eraw : 0;
    const float v   = live ? vals[e] : 0.0f;
    const int mycol = cols[e];
    const int myrow = rows[e];

    // Gather base pointers: vgpr vg of B-blk holds K = vg (+2 hi half) -> edge 4*blk+...
    // dc only contributes dc*64 bytes, folded into the load's immediate offset.
    const float* pb[4][2];
#pragma unroll
    for (int blk = 0; blk < 4; ++blk)
#pragma unroll
        for (int vg = 0; vg < 2; ++vg) {
            int c = __shfl(mycol, 4 * blk + vg + (hi ? 2 : 0), 32);
            pb[blk][vg] = x + (size_t)c * DIM + m;
        }

    // Scatter base pointers: D VGPR r holds edge r (lo half) / r+8 (hi half)
    float* py[8];
#pragma unroll
    for (int r = 0; r < 8; ++r) {
        int rr = __shfl(myrow, r + (hi ? 8 : 0), 32);
        py[r] = y + (size_t)rr * DIM + m;
    }

    // A = 16x4 slices of diag(val); independent of dc, build once.
    const int k0 = hi ? 2 : 0;
    const int k1 = hi ? 3 : 1;
    v2f amat[4];
#pragma unroll
    for (int blk = 0; blk < 4; ++blk) {
        amat[blk].x = (m == 4 * blk + k0) ? v : 0.0f;
        amat[blk].y = (m == 4 * blk + k1) ? v : 0.0f;
    }

#pragma unroll
    for (int dc = 0; dc < 4; ++dc) {       // 4 chunks of 16 dims
        v8f d = {};                        // C = 0, accumulate via chained WMMA
#pragma unroll
        for (int blk = 0; blk < 4; ++blk) {
            v2f b;                         // B = 4x16 slice of gathered rows
            b.x = pb[blk][0][dc * 16];
            b.y = pb[blk][1][dc * 16];
            d = __builtin_amdgcn_wmma_f32_16x16x4_f32(
                    false, amat[blk], false, b, (short)0, d, false, false);
        }
        // scatter-add: 8 rows per lane, dim = dc*16 + m
#pragma unroll
        for (int r = 0; r < 8; ++r)
            unsafeAtomicAdd(py[r] + dc * 16, d[r]);
    }
}

extern "C" void kernel_launch(void* const* d_in, const int* in_sizes, int n_in,
                              void* d_out, int out_size, void* d_ws, size_t ws_size,
                              hipStream_t stream)
{
    const float* user = (const float*)d_in[0];
    const float* item = (const float*)d_in[1];
    const int*   rows = (const int*)d_in[2];
    const int*   cols = (const int*)d_in[3];
    const float* vals = (const float*)d_in[4];
    float*       out  = (float*)d_out;

    float* buf0 = (float*)d_ws;                     // ping
    float* buf1 = buf0 + (size_t)NTOT * DIM;        // pong (needs 2*128MB ws)

    const int    tpb     = 256;                     // 8 wave32 per block
    const size_t totalv4 = (size_t)NTOT * DIM / 4;
    const int    eblocks = (int)((totalv4 + tpb - 1) / tpb);

    lgcn_init_kernel<<<eblocks, tpb, 0, stream>>>(user, item, buf0, buf1, out);

    const int ngroups = (NNZ_ + 15) / 16;
    const int wpb     = tpb / 32;
    const int sblocks = (ngroups + wpb - 1) / wpb;

    float* cur = buf0;
    float* nxt = buf1;
    for (int l = 0; l < NLAYERS; ++l) {
        lgcn_spmm_wmma_kernel<<<sblocks, tpb, 0, stream>>>(rows, cols, vals, cur, nxt, NNZ_);
        lgcn_addz_kernel<<<eblocks, tpb, 0, stream>>>(out, nxt, cur);  // also zeroes next target
        float* t = cur; cur = nxt; nxt = t;
    }
}